// PhotosynthesisQModel_53669911331256
// MI455X (gfx1250) — compile-verified
//
#include <hip/hip_runtime.h>
#include <hip/hip_bf16.h>

// ---------------------------------------------------------------------------
// PhotosynthesisQModel on MI455X (gfx1250, wave32)
//
// Phase 1 (1 workgroup): full GNN head -> shared86 vector -> fold the affine
//   lin5/lin6 shared part into a single scalar c_base (exact: no nonlinearity
//   between lin5 and lin6).
// Phase 2 (4096 workgroups): stream `actions` (164 MB = the only real traffic;
//   ~7 us floor at 23.3 TB/s, and it fits in the 192 MB L2 for replays).
//   Per wave: score 16 actions with V_WMMA_F32_16X16X4_F32 tiles:
//     x5 = act_tile[16x80] @ W5act[80x16]   (K padded 78->80 with zeros)
//     x6 = x5 @ W6 + c_base                 (wave32 shfl_xor reduction)
//   Staging uses b128 global loads (contiguous 16B-aligned tile span).
// ---------------------------------------------------------------------------

typedef float v2f __attribute__((ext_vector_type(2)));
typedef float v8f __attribute__((ext_vector_type(8)));

struct GnnParams {
  const float* W[6];  const float* b[6];
  const float* Ws[6]; const float* bs[6];   // pt1,pt2,pt3,si1,si2,si3
  const float* pt4W; const float* pt4b;
  const float* si4W; const float* si4b;
  const float* l3W;  const float* l3b;
  const float* l4W;  const float* l4b;
  const float* l5W;  const float* l5b;
  const float* l6W;  const float* l6b;
};

__device__ __forceinline__ float lrelu(float x) { return x > 0.f ? x : 0.01f * x; }

// ---------------------------------------------------------------------------
// Kernel 1: whole GNN + fold shared part of lin5/lin6 into one scalar.
// Tiny (37x16), single block of 256 threads; runtime is noise vs Phase 2.
// ---------------------------------------------------------------------------
__global__ __launch_bounds__(256) void gnn_head_kernel(
    const float* __restrict__ player_enc,
    const float* __restrict__ player_trees_enc,
    const float* __restrict__ tree_influence,
    const float* __restrict__ sun_influence,
    const float* __restrict__ sun_pos_enc,
    const float* __restrict__ player_suns,
    const float* __restrict__ remaining_turns,
    GnnParams P,
    float* __restrict__ cbase_out)
{
  __shared__ float A1[37][37], A2[37][37];
  __shared__ float X[37][16], AX[37][16], Y[37][16];
  __shared__ float H[74][8];
  __shared__ float rs1[37], rs2[37];
  __shared__ float sv[86];
  __shared__ float bias16[16];
  const int tid = threadIdx.x;

  // A_norm = (A + I) / rowsum(A + I)
  for (int idx = tid; idx < 37 * 37; idx += 256) {
    int i = idx / 37, j = idx % 37;
    float d = (i == j) ? 1.f : 0.f;
    A1[i][j] = tree_influence[idx] + d;
    A2[i][j] = sun_influence[idx] + d;
  }
  __syncthreads();
  if (tid < 37) {
    float s = 0.f;
    for (int j = 0; j < 37; ++j) s += A1[tid][j];
    rs1[tid] = s;
  } else if (tid >= 64 && tid < 64 + 37) {
    int i = tid - 64;
    float s = 0.f;
    for (int j = 0; j < 37; ++j) s += A2[i][j];
    rs2[i] = s;
  }
  __syncthreads();
  for (int idx = tid; idx < 37 * 37; idx += 256) {
    int i = idx / 37, j = idx % 37;
    A1[i][j] /= rs1[i];
    A2[i][j] /= rs2[i];
  }
  __syncthreads();

  // Two GCN branches (pt over A1, si over A2)
  for (int br = 0; br < 2; ++br) {
    for (int idx = tid; idx < 37 * 16; idx += 256)
      X[idx >> 4][idx & 15] = player_trees_enc[idx];
    __syncthreads();
    for (int l = 0; l < 3; ++l) {
      const float* W  = P.W[br * 3 + l];
      const float* b  = P.b[br * 3 + l];
      const float* Ws = P.Ws[br * 3 + l];
      const float* bs = P.bs[br * 3 + l];
      for (int idx = tid; idx < 37 * 16; idx += 256) {
        int i = idx >> 4, j = idx & 15;
        float t = 0.f;
        for (int k = 0; k < 37; ++k)
          t += (br == 0 ? A1[i][k] : A2[i][k]) * X[k][j];
        AX[i][j] = t;
      }
      __syncthreads();
      for (int idx = tid; idx < 37 * 16; idx += 256) {
        int i = idx >> 4, j = idx & 15;
        float t = b[j] + bs[j];
        for (int c = 0; c < 16; ++c)
          t += AX[i][c] * W[c * 16 + j] + X[i][c] * Ws[c * 16 + j];
        Y[i][j] = lrelu(t);
      }
      __syncthreads();
      for (int idx = tid; idx < 37 * 16; idx += 256)
        X[idx >> 4][idx & 15] = Y[idx >> 4][idx & 15];
      __syncthreads();
    }
    const float* W4 = (br == 0) ? P.pt4W : P.si4W;
    const float* b4 = (br == 0) ? P.pt4b : P.si4b;
    for (int idx = tid; idx < 37 * 8; idx += 256) {
      int i = idx >> 3, j = idx & 7;
      float t = b4[j];
      for (int c = 0; c < 16; ++c) t += X[i][c] * W4[c * 8 + j];
      H[br * 37 + i][j] = t;
    }
    __syncthreads();
  }

  // lin3 (lrelu) -> lin4 -> shared86 vector
  if (tid < 74) {
    float z[8];
    for (int j = 0; j < 8; ++j) {
      float t = P.l3b[j];
      for (int c = 0; c < 8; ++c) t += H[tid][c] * P.l3W[c * 8 + j];
      z[j] = lrelu(t);
    }
    float x4 = P.l4b[0];
    for (int c = 0; c < 8; ++c) x4 += z[c] * P.l4W[c];
    sv[tid] = x4;
  } else if (tid == 74) sv[74] = player_suns[0];
  else if (tid >= 75 && tid <= 80) sv[tid] = sun_pos_enc[tid - 75];
  else if (tid >= 81 && tid <= 84) sv[tid] = player_enc[tid - 81];
  else if (tid == 85) sv[85] = remaining_turns[0];
  __syncthreads();

  // bias16 = shared86 @ W5[:86] + b5 ; c_base = bias16 @ W6 + b6
  if (tid < 16) {
    float t = P.l5b[tid];
    for (int i = 0; i < 86; ++i) t += sv[i] * P.l5W[i * 16 + tid];
    bias16[tid] = t;
  }
  __syncthreads();
  if (tid == 0) {
    float t = P.l6b[0];
    for (int n = 0; n < 16; ++n) t += bias16[n] * P.l6W[n];
    cbase_out[0] = t;
  }
}

// ---------------------------------------------------------------------------
// Kernel 2: HBM-bound action scoring with V_WMMA_F32_16X16X4_F32.
// 256 threads = 8 wave32 waves; each wave scores 16 action rows.
// A fragment (16x4 f32): lane = M (lane&15), VGPR0 = K=2*half, VGPR1 = K=2*half+1.
// B fragment (4x16 f32): lane = N (lane&15), same K-per-VGPR split.
// C/D (16x16 f32): lane half selects M in {j, j+8}, N = lane&15, VGPR j = row.
// ---------------------------------------------------------------------------
__global__ __launch_bounds__(256) void score_actions_wmma(
    const float* __restrict__ actions,
    const float* __restrict__ l5W,   // [164,16]
    const float* __restrict__ l6W,   // [16,1]
    const float* __restrict__ cbase,
    float* __restrict__ out,
    long long Btot)
{
  __shared__ float tile[8][16][80];  // per-wave 16 rows x K=80 (78 + zero pad)
  __shared__ float Wb[80][16];       // W5 action slice, zero-padded rows 78,79
  __shared__ float w6s[16];
  __shared__ float cb;

  const int tid  = threadIdx.x;
  const int w    = tid >> 5;
  const int lane = tid & 31;

  // Stage shared operands (block-wide; 5 KB, L2-resident after first block)
  for (int i = tid; i < 78 * 16; i += 256)
    Wb[i >> 4][i & 15] = l5W[(86 + (i >> 4)) * 16 + (i & 15)];
  for (int i = tid; i < 2 * 16; i += 256)
    Wb[78 + (i >> 4)][i & 15] = 0.f;
  if (tid < 16) w6s[tid] = l6W[tid];
  if (tid == 0) cb = cbase[0];

  // Stage this wave's 16x78 action tile. Rows are contiguous in memory, so the
  // tile is one contiguous, 16B-aligned 4992 B span (row0 % 16 == 0):
  // use b128 global loads (312 float4s), repad 78->80 on the LDS-store side.
  const long long row0 = (long long)blockIdx.x * 128 + (long long)w * 16;
  const float* src = actions + row0 * 78;

  if (row0 + 16 <= Btot) {
    const float4* src4 = (const float4*)src;
#pragma unroll 2
    for (int v = lane; v < (16 * 78) / 4; v += 32) {   // 312 vec4s, ~10 iters
      float4 d = src4[v];
      int idx = 4 * v;
      int r = idx / 78;
      int c = idx - r * 78;
      float e0 = d.x, e1 = d.y, e2 = d.z, e3 = d.w;
      tile[w][r][c] = e0; if (++c == 78) { c = 0; ++r; }
      tile[w][r][c] = e1; if (++c == 78) { c = 0; ++r; }
      tile[w][r][c] = e2; if (++c == 78) { c = 0; ++r; }
      tile[w][r][c] = e3;
    }
  } else {
    // Defensive tail path (B is a multiple of 128 in practice)
    for (int i = lane; i < 16 * 78; i += 32) {
      int r = i / 78, c = i - r * 78;
      tile[w][r][c] = (row0 + r < Btot) ? src[i] : 0.f;
    }
  }
  { // zero the 2 pad columns: 16 rows x 2 cols == 32 lanes, one store each
    int r = lane >> 1, c = 78 + (lane & 1);
    tile[w][r][c] = 0.f;
  }
  __syncthreads();

  const int n  = lane & 15;   // N column (also M row for the A fragment)
  const int kh = lane >> 4;   // lane-half -> K sub-offset

  v8f c = {0.f, 0.f, 0.f, 0.f, 0.f, 0.f, 0.f, 0.f};
#pragma unroll
  for (int k = 0; k < 80; k += 4) {
    v2f a, b;
    a.x = tile[w][n][k + 2 * kh];
    a.y = tile[w][n][k + 2 * kh + 1];
    b.x = Wb[k + 2 * kh][n];
    b.y = Wb[k + 2 * kh + 1][n];
    c = __builtin_amdgcn_wmma_f32_16x16x4_f32(
        /*neg_a=*/false, a, /*neg_b=*/false, b,
        /*c_mod=*/(short)0, c, /*reuse_a=*/false, /*reuse_b=*/false);
  }

  // x6[m] = sum_n x5[m][n] * W6[n] + c_base  (reduce across the 16-lane half)
  const float w6n = w6s[n];
  const float cbv = cb;
#pragma unroll
  for (int j = 0; j < 8; ++j) {
    float t = c[j] * w6n;
    t += __shfl_xor(t, 8, 32);
    t += __shfl_xor(t, 4, 32);
    t += __shfl_xor(t, 2, 32);
    t += __shfl_xor(t, 1, 32);
    if (n == 0) {
      long long row = row0 + j + 8 * kh;
      if (row < Btot) out[row] = t + cbv;
    }
  }
}

// ---------------------------------------------------------------------------
// Launch. Input order (setup_inputs insertion order, params flattened
// depth-first in insertion order): 0..7 tensors, then 36 param arrays.
// ---------------------------------------------------------------------------
extern "C" void kernel_launch(void* const* d_in, const int* in_sizes, int n_in,
                              void* d_out, int out_size, void* d_ws, size_t ws_size,
                              hipStream_t stream) {
  const float* player_enc      = (const float*)d_in[0];
  const float* player_trees    = (const float*)d_in[1];
  const float* tree_influence  = (const float*)d_in[2];
  const float* sun_influence   = (const float*)d_in[3];
  const float* sun_pos_enc     = (const float*)d_in[4];
  const float* player_suns     = (const float*)d_in[5];
  const float* remaining_turns = (const float*)d_in[6];
  const float* actions         = (const float*)d_in[7];

  GnnParams P;
  int base = 8;
  for (int i = 0; i < 6; ++i) {  // pt1, pt2, pt3, si1, si2, si3
    P.W[i]  = (const float*)d_in[base + 0];
    P.b[i]  = (const float*)d_in[base + 1];
    P.Ws[i] = (const float*)d_in[base + 2];
    P.bs[i] = (const float*)d_in[base + 3];
    base += 4;
  }
  P.pt4W = (const float*)d_in[base + 0];
  P.pt4b = (const float*)d_in[base + 1];
  P.si4W = (const float*)d_in[base + 2];
  P.si4b = (const float*)d_in[base + 3];
  P.l3W  = (const float*)d_in[base + 4];
  P.l3b  = (const float*)d_in[base + 5];
  P.l4W  = (const float*)d_in[base + 6];
  P.l4b  = (const float*)d_in[base + 7];
  P.l5W  = (const float*)d_in[base + 8];
  P.l5b  = (const float*)d_in[base + 9];
  P.l6W  = (const float*)d_in[base + 10];
  P.l6b  = (const float*)d_in[base + 11];

  float* cbase = (float*)d_ws;

  gnn_head_kernel<<<1, 256, 0, stream>>>(
      player_enc, player_trees, tree_influence, sun_influence,
      sun_pos_enc, player_suns, remaining_turns, P, cbase);

  const long long B = (long long)in_sizes[7] / 78;
  const int nblocks = (int)((B + 127) / 128);
  score_actions_wmma<<<nblocks, 256, 0, stream>>>(
      actions, P.l5W, P.l6W, cbase, (float*)d_out, B);
}